// AvgSeq_57921928954377
// MI455X (gfx1250) — compile-verified
//
#include <hip/hip_runtime.h>
#include <hip/hip_bf16.h>

// Problem: out[b,s,d] = cumsum_s(x)[b,s,d] / (s+1), x: [16, 8192, 256] f32.
// Memory-bound (268 MB min traffic ~ 11.5us @ 23.3 TB/s). Strategy:
//  pass1: per-chunk (512-seq) column sums          (read 128MiB, write 256KB)
//  pass2: exclusive scan of chunk sums per column  (tiny)
//  pass3: WMMA-based intra-chunk scan: Y = L*X + prefix with L = 16x16
//         lower-triangular ones via 4x V_WMMA_F32_16X16X4_F32 (exact in f32),
//         then multiply by NR-refined 1/(s+1) and store.
// 4096 independent wave tasks in pass3 hide the serial per-tile dependency.
// Division replaced by v_rcp_f32 + 1 Newton step (~1 ulp; divisor is an
// exact-in-f32 integer <= 8192) to keep pass3 bandwidth-bound, not VALU-bound.

typedef float v2f __attribute__((ext_vector_type(2)));
typedef float v8f __attribute__((ext_vector_type(8)));

constexpr int BB = 16;               // batch
constexpr int SS = 8192;             // seq
constexpr int DD = 256;              // channels
constexpr int CHUNKS = 16;           // seq chunks
constexpr int CS = SS / CHUNKS;      // 512 seq per chunk
constexpr int TILES = CS / 16;       // 32 16-row tiles per chunk

// ---------------- Pass 1: per-chunk column sums --------------------------
__global__ void __launch_bounds__(256)
avgseq_pass1_chunk_sums(const float* __restrict__ in, float* __restrict__ sums) {
    const int b = blockIdx.x / CHUNKS;
    const int c = blockIdx.x % CHUNKS;
    const int d = threadIdx.x;                       // 0..255, coalesced
    const float* p = in + ((size_t)(b * SS + c * CS) * DD + d);
    float acc = 0.0f;
#pragma unroll 8
    for (int s = 0; s < CS; ++s) acc += p[(size_t)s * DD];
    sums[(b * CHUNKS + c) * DD + d] = acc;
}

// ---------------- Pass 2: exclusive scan of chunk sums (in place) --------
__global__ void __launch_bounds__(256)
avgseq_pass2_scan_sums(float* __restrict__ sums) {
    const int tid = blockIdx.x * blockDim.x + threadIdx.x;   // 0..4095
    const int b = tid / DD;
    const int d = tid % DD;
    float run = 0.0f;
#pragma unroll
    for (int c = 0; c < CHUNKS; ++c) {
        const int idx = (b * CHUNKS + c) * DD + d;
        const float t = sums[idx];
        sums[idx] = run;
        run += t;
    }
}

// Fast ~1ulp reciprocal: v_rcp_f32 + one Newton-Raphson step.
__device__ __forceinline__ float fast_rcp(float den) {
    float r = __builtin_amdgcn_rcpf(den);
    return __builtin_fmaf(__builtin_fmaf(-den, r, 1.0f), r, r);
}

// ---------------- Pass 3: WMMA tile scan + divide ------------------------
// One wave per (batch, d-tile of 16, chunk) = 4096 wave tasks, 8 waves/block.
__global__ void __launch_bounds__(256)
avgseq_pass3_wmma_scan(const float* __restrict__ in,
                       const float* __restrict__ prefixes,
                       float* __restrict__ out) {
    const int lane  = threadIdx.x & 31;
    const int wave  = threadIdx.x >> 5;
    const int task  = blockIdx.x * 8 + wave;     // 0..4095
    const int b     = task >> 8;                 // 16
    const int dtile = (task >> 4) & 15;          // 16
    const int chunk = task & 15;                 // 16

    const int  n  = lane & 15;                   // column within d-tile
    const bool hi = lane >= 16;
    const int  d0 = dtile * 16;

    // A = 16x16 lower-triangular ones, fed as 4 K-slices of 16x4 f32.
    // A-layout (16x4): lanes 0-15 & 16-31 both hold M=lane&15;
    // VGPR0 = K {0 | 2}, VGPR1 = K {1 | 3} (low|high half-wave).
    const int m     = n;
    const int kbase = hi ? 2 : 0;
    v2f A[4];
#pragma unroll
    for (int kk = 0; kk < 4; ++kk) {
        const int k0 = kk * 4 + kbase;
        A[kk].x = (k0     <= m) ? 1.0f : 0.0f;
        A[kk].y = (k0 + 1 <= m) ? 1.0f : 0.0f;
    }

    // Running column prefix for this chunk (same value in lane and lane+16).
    float prefix = prefixes[(b * CHUNKS + chunk) * DD + d0 + n];

    const size_t base = ((size_t)b * SS + (size_t)chunk * CS) * DD + d0 + n;
    const float* pin  = in  + base;
    float*       pout = out + base;

    // B-layout (4x16): lanes 0-15 hold rows {k,k+1}, lanes 16-31 rows {k+2,k+3}.
    const int rowx0 = hi ? 2 : 0;
    const int mrow0 = hi ? 8 : 0;   // C/D: VGPR r holds M=r (lo) / M=r+8 (hi)

    // Denominators (s+1) as exact float counters; advanced by +16.0f per tile
    // (all values <= 8192, exactly representable -> no per-tile cvt).
    float denom[8];
#pragma unroll
    for (int r = 0; r < 8; ++r)
        denom[r] = (float)(chunk * CS + r + mrow0 + 1);

    for (int t = 0; t < TILES; ++t) {
        const size_t toff = (size_t)t * 16 * DD;

        v2f   Bm[4];
        float partial = 0.0f;       // this lane's share of the tile column sum
#pragma unroll
        for (int kk = 0; kk < 4; ++kk) {
            const int   rx = kk * 4 + rowx0;
            const float bx = pin[toff + (size_t)rx * DD];
            const float by = pin[toff + (size_t)(rx + 1) * DD];
            Bm[kk].x = bx;
            Bm[kk].y = by;
            partial += bx + by;
        }

        // Prefetch next tile's rows for this lane (global_prefetch_b8).
        if (t + 1 < TILES) {
            __builtin_prefetch(&pin[toff + (size_t)(16 + rowx0) * DD], 0, 0);
            __builtin_prefetch(&pin[toff + (size_t)(24 + rowx0) * DD], 0, 0);
        }

        // C = prefix broadcast across all 16 output rows.
        v8f c;
#pragma unroll
        for (int i = 0; i < 8; ++i) c[i] = prefix;

        // Y = L * X + prefix, K=16 as 4 chained 16x16x4 f32 WMMAs.
#pragma unroll
        for (int kk = 0; kk < 4; ++kk) {
            c = __builtin_amdgcn_wmma_f32_16x16x4_f32(
                /*neg_a=*/false, A[kk],
                /*neg_b=*/false, Bm[kk],
                /*c_mod=*/(short)0, c,
                /*reuse_a=*/false, /*reuse_b=*/false);
        }

        // New running prefix: add full 16-row column sum (partner holds the
        // other 8 rows of column n at lane^16).
        prefix += partial + __shfl_xor(partial, 16, 32);

        // Multiply by refined 1/(s+1) and store.
#pragma unroll
        for (int r = 0; r < 8; ++r) {
            pout[toff + (size_t)(r + mrow0) * DD] = c[r] * fast_rcp(denom[r]);
            denom[r] += 16.0f;
        }
    }
}

extern "C" void kernel_launch(void* const* d_in, const int* in_sizes, int n_in,
                              void* d_out, int out_size, void* d_ws, size_t ws_size,
                              hipStream_t stream) {
    const float* in  = (const float*)d_in[0];
    float*       out = (float*)d_out;
    float*       sums = (float*)d_ws;   // 16*16*256 floats = 256 KiB

    // Pass 1: 256 blocks x 256 threads
    avgseq_pass1_chunk_sums<<<BB * CHUNKS, DD, 0, stream>>>(in, sums);
    // Pass 2: 4096 threads
    avgseq_pass2_scan_sums<<<(BB * DD) / 256, 256, 0, stream>>>(sums);
    // Pass 3: 4096 wave-tasks, 8 waves (256 threads) per block
    avgseq_pass3_wmma_scan<<<(BB * 16 * CHUNKS) / 8, 256, 0, stream>>>(in, sums, out);
}